// Conv1d_NN_44976897523804
// MI455X (gfx1250) — compile-verified
//
#include <hip/hip_runtime.h>

// ---------------------------------------------------------------------------
// Conv1d_NN for MI455X (gfx1250): KNN via FP32 WMMA distance GEMM + fused
// top-3 selection in LDS, then gathered conv1d as a second WMMA GEMM.
// ---------------------------------------------------------------------------

typedef __attribute__((ext_vector_type(2))) float v2f;
typedef __attribute__((ext_vector_type(8))) float v8f;

#define BATCH 16
#define CH    64
#define TLEN  2048
#define KNN   3

// ---------------------------------------------------------------------------
// Kernel 1: per-token squared norms  nsq[b,t] = sum_c x[b,c,t]^2
// ---------------------------------------------------------------------------
__global__ __launch_bounds__(256) void nsq_kernel(const float* __restrict__ x,
                                                  float* __restrict__ nsq) {
    int g = blockIdx.x * 256 + threadIdx.x;          // over B*T
    int b = g >> 11;
    int t = g & (TLEN - 1);
    const float* xb = x + (size_t)b * CH * TLEN + t;
    float s = 0.0f;
#pragma unroll
    for (int c = 0; c < CH; ++c) {
        float v = xb[c * TLEN];
        s += v * v;
    }
    nsq[g] = s;
}

// ---------------------------------------------------------------------------
// Kernel 2: distances for a 16-token row tile vs all 2048 tokens via
// V_WMMA_F32_16X16X4_F32, slab kept in LDS, fused top-3 selection.
// Block: 256 threads (8 waves). Wave w handles column chunks s0=(it*8+w)*16.
// Dynamic LDS: dist[16][2048] + cand_d[16][48] + cand_i[16][48]  (~137 KB)
// ---------------------------------------------------------------------------
__global__ __launch_bounds__(256) void knn_kernel(const float* __restrict__ x,
                                                  const float* __restrict__ nsq,
                                                  int* __restrict__ idx_out) {
    const int b    = blockIdx.x >> 7;                // T/16 = 128 tiles/batch
    const int tile = blockIdx.x & 127;
    const int t0   = tile * 16;

    extern __shared__ float smem[];
    float* dist   = smem;                             // [16][2048]
    float* cand_d = smem + 16 * TLEN;                 // [16][48]
    int*   cand_i = (int*)(cand_d + 16 * 48);         // [16][48]

    const int tid   = threadIdx.x;
    const int lane  = tid & 31;
    const int wave  = tid >> 5;                       // 0..7
    const int lmod  = lane & 15;
    const int lhalf = (lane < 16) ? 0 : 2;            // K offset for A/B frags
    const int mhalf = (lane < 16) ? 0 : 8;            // M offset for C/D

    const float* xb   = x   + (size_t)b * CH * TLEN;
    const float* nsqb = nsq + (size_t)b * TLEN;

    // Preload all A fragments for this 16-row tile (reused by every s-chunk).
    // A 16x4 f32 layout: lane = M (mod 16); vgpr0 holds K = base+lhalf,
    // vgpr1 holds K = base+lhalf+1.
    v2f afrag[16];
#pragma unroll
    for (int kc = 0; kc < 16; ++kc) {
        int c0 = kc * 4 + lhalf;
        afrag[kc].x = xb[(size_t)(c0 + 0) * TLEN + t0 + lmod];
        afrag[kc].y = xb[(size_t)(c0 + 1) * TLEN + t0 + lmod];
    }
    float nsq_t[8];
#pragma unroll
    for (int r = 0; r < 8; ++r) nsq_t[r] = nsqb[t0 + r + mhalf];

    // 16 iterations x 8 waves x 16 columns = 2048 columns
    for (int it = 0; it < 16; ++it) {
        const int s0 = (it * 8 + wave) * 16;
        v8f acc = {};
#pragma unroll
        for (int kc = 0; kc < 16; ++kc) {
            int c0 = kc * 4 + lhalf;
            v2f bfrag;
            bfrag.x = xb[(size_t)(c0 + 0) * TLEN + s0 + lmod];
            bfrag.y = xb[(size_t)(c0 + 1) * TLEN + s0 + lmod];
            acc = __builtin_amdgcn_wmma_f32_16x16x4_f32(
                false, afrag[kc], false, bfrag, (short)0, acc, false, false);
        }
        const float nsq_s = nsqb[s0 + lmod];
#pragma unroll
        for (int r = 0; r < 8; ++r) {
            int m   = r + mhalf;                       // row within tile
            float d = nsq_t[r] + nsq_s - 2.0f * acc[r];
            dist[m * TLEN + s0 + lmod] = fmaxf(d, 0.0f);
        }
    }
    __syncthreads();

    // --- Top-3 selection: 16 threads per row, each scans a strided slice ---
    const int row = tid >> 4;                          // 0..15
    const int tt  = tid & 15;
    float d0 = 3.4e38f, d1 = 3.4e38f, d2 = 3.4e38f;
    int   i0 = 0x7fffffff, i1 = 0x7fffffff, i2 = 0x7fffffff;
    const float* drow = dist + row * TLEN;
    for (int j = tt; j < TLEN; j += 16) {
        float d = drow[j];
        if (d < d0)      { d2 = d1; i2 = i1; d1 = d0; i1 = i0; d0 = d; i0 = j; }
        else if (d < d1) { d2 = d1; i2 = i1; d1 = d;  i1 = j; }
        else if (d < d2) { d2 = d;  i2 = j; }
    }
    cand_d[row * 48 + tt * 3 + 0] = d0;  cand_i[row * 48 + tt * 3 + 0] = i0;
    cand_d[row * 48 + tt * 3 + 1] = d1;  cand_i[row * 48 + tt * 3 + 1] = i1;
    cand_d[row * 48 + tt * 3 + 2] = d2;  cand_i[row * 48 + tt * 3 + 2] = i2;
    __syncthreads();

    if (tt == 0) {
        // Merge 48 candidates with (dist, index) lexicographic order to match
        // lax.top_k stable tie-breaking (lower index first).
        float bd0 = 3.4e38f, bd1 = 3.4e38f, bd2 = 3.4e38f;
        int   bi0 = 0x7fffffff, bi1 = 0x7fffffff, bi2 = 0x7fffffff;
        for (int c = 0; c < 48; ++c) {
            float d = cand_d[row * 48 + c];
            int   i = cand_i[row * 48 + c];
            if (d < bd0 || (d == bd0 && i < bi0)) {
                bd2 = bd1; bi2 = bi1; bd1 = bd0; bi1 = bi0; bd0 = d; bi0 = i;
            } else if (d < bd1 || (d == bd1 && i < bi1)) {
                bd2 = bd1; bi2 = bi1; bd1 = d; bi1 = i;
            } else if (d < bd2 || (d == bd2 && i < bi2)) {
                bd2 = d; bi2 = i;
            }
        }
        int* o = idx_out + ((size_t)b * TLEN + t0 + row) * KNN;
        o[0] = bi0; o[1] = bi1; o[2] = bi2;
    }
}

// ---------------------------------------------------------------------------
// Kernel 3: out[b,o,t] = bias[o] + sum_{i,kk} W[o,i,kk] * x[b,i,idx[b,t,kk]]
// = (64 x 192) W-matrix times gathered (192 x 16) token tile, via WMMA f32.
// Block: 256 threads (8 waves): wave -> (o-tile 0..3, token-subtile 0..1).
// ---------------------------------------------------------------------------
__global__ __launch_bounds__(256) void conv_kernel(const float* __restrict__ x,
                                                   const float* __restrict__ W,
                                                   const float* __restrict__ bias,
                                                   const int* __restrict__ idx,
                                                   float* __restrict__ out) {
    const int b   = blockIdx.x >> 6;                 // T/32 = 64 blocks/batch
    const int blk = blockIdx.x & 63;
    const int wave  = threadIdx.x >> 5;
    const int lane  = threadIdx.x & 31;
    const int lmod  = lane & 15;
    const int lhalf = (lane < 16) ? 0 : 2;
    const int mhalf = (lane < 16) ? 0 : 8;

    const int o0 = (wave & 3) * 16;                  // output-channel tile
    const int t0 = blk * 32 + (wave >> 2) * 16;      // token tile
    const int tok = t0 + lmod;

    const float* xb = x + (size_t)b * CH * TLEN;
    const int* my = idx + ((size_t)b * TLEN + tok) * KNN;
    const int s0i = my[0], s1i = my[1], s2i = my[2];

    v8f acc = {};
#pragma unroll
    for (int kc = 0; kc < 48; ++kc) {                // K = 192 = 48 * 4
        const int ka0 = kc * 4 + lhalf;              // flattened k = i*3 + kk
        const int ka1 = ka0 + 1;
        v2f a, bb;
        // A = W reshaped (O, 192); contiguous in k.
        a.x = W[(size_t)(o0 + lmod) * (CH * KNN) + ka0];
        a.y = W[(size_t)(o0 + lmod) * (CH * KNN) + ka1];
        // B = gathered neighbor features.
        {
            int i = ka0 / 3, kk = ka0 - i * 3;
            int s = (kk == 0) ? s0i : ((kk == 1) ? s1i : s2i);
            bb.x = xb[(size_t)i * TLEN + s];
        }
        {
            int i = ka1 / 3, kk = ka1 - i * 3;
            int s = (kk == 0) ? s0i : ((kk == 1) ? s1i : s2i);
            bb.y = xb[(size_t)i * TLEN + s];
        }
        acc = __builtin_amdgcn_wmma_f32_16x16x4_f32(
            false, a, false, bb, (short)0, acc, false, false);
    }

#pragma unroll
    for (int r = 0; r < 8; ++r) {
        int o = o0 + r + mhalf;
        out[(size_t)b * CH * TLEN + (size_t)o * TLEN + tok] = acc[r] + bias[o];
    }
}

// ---------------------------------------------------------------------------
// Launch
// ---------------------------------------------------------------------------
extern "C" void kernel_launch(void* const* d_in, const int* in_sizes, int n_in,
                              void* d_out, int out_size, void* d_ws, size_t ws_size,
                              hipStream_t stream) {
    (void)in_sizes; (void)n_in; (void)out_size; (void)ws_size;
    const float* x    = (const float*)d_in[0];   // (16, 64, 2048)
    const float* W    = (const float*)d_in[1];   // (64, 64, 3)
    const float* bias = (const float*)d_in[2];   // (64,)
    float* out = (float*)d_out;                  // (16, 64, 2048)

    float* nsq = (float*)d_ws;                                   // B*T floats
    int*   idx = (int*)((char*)d_ws + (size_t)BATCH * TLEN * sizeof(float));

    nsq_kernel<<<(BATCH * TLEN) / 256, 256, 0, stream>>>(x, nsq);

    const size_t smem = (size_t)16 * TLEN * sizeof(float)   // dist slab 128 KB
                      + (size_t)16 * 48 * sizeof(float)     // cand dists
                      + (size_t)16 * 48 * sizeof(int);      // cand indices
    // WGP has 320 KB LDS; raise the per-kernel dynamic-LDS cap above 64 KB.
    hipFuncSetAttribute((const void*)knn_kernel,
                        hipFuncAttributeMaxDynamicSharedMemorySize, (int)smem);
    knn_kernel<<<BATCH * (TLEN / 16), 256, smem, stream>>>(x, nsq, idx);

    conv_kernel<<<BATCH * (TLEN / 32), 256, 0, stream>>>(x, W, bias, idx, out);
}